// SelfAttentionWithKeyFCLayer_85160611545290
// MI455X (gfx1250) — compile-verified
//
#include <hip/hip_runtime.h>
#include <hip/hip_bf16.h>

// MI455X / gfx1250, wave32. f32 WMMA (16x16x4) used for all GEMMs.

typedef __attribute__((ext_vector_type(2))) float v2f;
typedef __attribute__((ext_vector_type(8))) float v8f;

#define BATCH 16
#define SEQ   2048
#define CH    128   // C == W == 128

static __device__ __forceinline__ v8f wmma4(v2f a, v2f b, v8f c) {
    // 8 args: (neg_a, A, neg_b, B, c_mod, C, reuse_a, reuse_b)
    return __builtin_amdgcn_wmma_f32_16x16x4_f32(false, a, false, b, (short)0, c, false, false);
}

// ---------------------------------------------------------------------------
// Weight collapse: y = ((x W0^T + b0) W1^T + b1) W2^T + b2
//   => Weff = W2 W1 W0 ; beff = W2 (W1 b0 + b1) + b2
// Stored transposed: Wt[k][n] = Weff[n][k] so out = X @ Wt + beff.
// ---------------------------------------------------------------------------
__global__ void collapse1_kernel(const float* qw, const float* kw, const float* vw,
                                 const float* qb, const float* kb, const float* vb,
                                 float* T, float* tb) {
    int mat = blockIdx.y;
    const float* Wm = (mat == 0) ? qw : (mat == 1) ? kw : vw;   // [3][128][128]
    const float* bm = (mat == 0) ? qb : (mat == 1) ? kb : vb;   // [3][128]
    const float* W0 = Wm;
    const float* W1 = Wm + CH * CH;
    int i = blockIdx.x;      // row of W1
    int j = threadIdx.x;     // col of W0
    float s = 0.f;
    for (int l = 0; l < CH; ++l) s += W1[i * CH + l] * W0[l * CH + j];
    T[mat * CH * CH + i * CH + j] = s;                 // T = W1 W0
    if (blockIdx.x == 0) {
        float t = 0.f;
        for (int l = 0; l < CH; ++l) t += W1[j * CH + l] * bm[l];
        tb[mat * CH + j] = t + bm[CH + j];             // tb = W1 b0 + b1
    }
}

__global__ void collapse2_kernel(const float* qw, const float* kw, const float* vw,
                                 const float* qb, const float* kb, const float* vb,
                                 const float* T, const float* tb,
                                 float* Wt, float* beff) {
    int mat = blockIdx.y;
    const float* Wm = (mat == 0) ? qw : (mat == 1) ? kw : vw;
    const float* bm = (mat == 0) ? qb : (mat == 1) ? kb : vb;
    const float* W2 = Wm + 2 * CH * CH;
    int i = blockIdx.x;      // out-channel of Weff
    int j = threadIdx.x;     // in-channel
    float s = 0.f;
    for (int l = 0; l < CH; ++l) s += W2[i * CH + l] * T[mat * CH * CH + l * CH + j];
    Wt[mat * CH * CH + j * CH + i] = s;                // transposed store
    if (blockIdx.x == 0) {
        float t = 0.f;
        for (int l = 0; l < CH; ++l) t += W2[j * CH + l] * tb[mat * CH + l];
        beff[mat * CH + j] = t + bm[2 * CH + j];
    }
}

// ---------------------------------------------------------------------------
// Projection: Out[row][n] = sum_k X[row][k] * Wt[k][n] + beff[n]
// 1 wave = 16 rows x 128 cols, K=128 chained in 16x16x4 WMMAs.
// ---------------------------------------------------------------------------
__global__ __launch_bounds__(128) void proj_kernel(const float* __restrict__ X,
                                                   const float* __restrict__ Wt,
                                                   const float* __restrict__ beff,
                                                   float* __restrict__ Out) {
    int lane = threadIdx.x & 31;
    int wave = threadIdx.x >> 5;
    int r0   = (blockIdx.x * 4 + wave) * 16;
    int lr   = lane & 15;
    int hi   = lane >> 4;

    const float* xrow = X + (size_t)(r0 + lr) * CH;
    v2f a[32];
#pragma unroll
    for (int kk = 0; kk < 32; ++kk)
        a[kk] = *(const v2f*)(xrow + 4 * kk + 2 * hi);

#pragma unroll
    for (int nt = 0; nt < 8; ++nt) {
        int n0 = nt * 16;
        v8f c = {};
#pragma unroll
        for (int kk = 0; kk < 32; ++kk) {
            int k0 = 4 * kk + 2 * hi;
            v2f b;
            b.x = Wt[(k0 + 0) * CH + n0 + lr];
            b.y = Wt[(k0 + 1) * CH + n0 + lr];
            c = wmma4(a[kk], b, c);
        }
        float bias = beff[n0 + lr];
#pragma unroll
        for (int r = 0; r < 8; ++r)
            Out[(size_t)(r0 + r + 8 * hi) * CH + n0 + lr] = c[r] + bias;
    }
}

// ---------------------------------------------------------------------------
// Flash-style attention + masked softmax + LayerNorm + residual.
// 1 wave = 16 query rows of one batch; two passes over M (stats, then PV).
// ---------------------------------------------------------------------------
__global__ __launch_bounds__(128) void attn_kernel(
        const float* __restrict__ Q, const float* __restrict__ Kd,
        const float* __restrict__ Vd, const unsigned char* __restrict__ mask,
        const float* __restrict__ x1, const float* __restrict__ ln_g,
        const float* __restrict__ ln_b, float* __restrict__ Out) {
    __shared__ float pstage[4 * 16 * 17];   // per-wave 16x16 P tile, padded stride 17

    int lane = threadIdx.x & 31;
    int wave = threadIdx.x >> 5;
    int b    = blockIdx.y;
    int r0   = (blockIdx.x * 4 + wave) * 16;      // query row base within batch
    int lr   = lane & 15;
    int hi   = lane >> 4;

    const float scale = 0.088388347648318447f;    // 1/sqrt(128)
    const float* qbase = Q + ((size_t)b * SEQ + r0 + lr) * CH;
    v2f qa[32];
#pragma unroll
    for (int kk = 0; kk < 32; ++kk) {
        v2f q = *(const v2f*)(qbase + 4 * kk + 2 * hi);
        qa[kk] = q * scale;
    }

    const float* kbase = Kd + (size_t)b * SEQ * CH;
    const float* vbat  = Vd + (size_t)b * SEQ * CH;
    const unsigned char* mbase = mask + ((size_t)b * SEQ + r0) * (size_t)SEQ;

    // ---- Pass 1: per-lane online max / sum-of-exp (masked -> s=-1e12, p=0) ----
    float pmax[8], psum[8];
#pragma unroll
    for (int r = 0; r < 8; ++r) { pmax[r] = -3.0e38f; psum[r] = 0.f; }

    for (int mt = 0; mt < 128; ++mt) {
        int m0 = mt * 16;
        const float* krow = kbase + (size_t)(m0 + lr) * CH;
        v8f c = {};
#pragma unroll
        for (int kk = 0; kk < 32; ++kk) {
            v2f kb = *(const v2f*)(krow + 4 * kk + 2 * hi);
            c = wmma4(qa[kk], kb, c);
        }
#pragma unroll
        for (int r = 0; r < 8; ++r) {
            unsigned char mk = mbase[(size_t)(r + 8 * hi) * SEQ + m0 + lr];
            float sv = mk ? -1.0e12f : c[r];
            float nm = fmaxf(pmax[r], sv);
            float pv = mk ? 0.f : __expf(sv - nm);
            psum[r] = psum[r] * __expf(pmax[r] - nm) + pv;
            pmax[r] = nm;
        }
    }

    // ---- cross-lane reduction within each 16-lane half (wave32) ----
    float rowmax[8], rowinv[8];
#pragma unroll
    for (int r = 0; r < 8; ++r) {
        float mmax = pmax[r], msum = psum[r];
#pragma unroll
        for (int d = 1; d < 16; d <<= 1) {
            float omax = __shfl_xor(mmax, d);
            float osum = __shfl_xor(msum, d);
            float nm = fmaxf(mmax, omax);
            msum = msum * __expf(mmax - nm) + osum * __expf(omax - nm);
            mmax = nm;
        }
        rowmax[r] = mmax;
        rowinv[r] = (msum > 0.f) ? (1.f / msum) : 0.f;   // all-masked row -> 0
    }

    // ---- Pass 2: recompute scores, P = exp(s-max) (masked->0), acc += P @ V ----
    v8f acc[8];
#pragma unroll
    for (int nt = 0; nt < 8; ++nt) acc[nt] = (v8f){};

    float* st = pstage + wave * 16 * 17;

    for (int mt = 0; mt < 128; ++mt) {
        int m0 = mt * 16;
        const float* krow = kbase + (size_t)(m0 + lr) * CH;
        v8f c = {};
#pragma unroll
        for (int kk = 0; kk < 32; ++kk) {
            v2f kb = *(const v2f*)(krow + 4 * kk + 2 * hi);
            c = wmma4(qa[kk], kb, c);
        }
        // stage P tile to LDS in D-layout, read back in A-layout (wave-private)
#pragma unroll
        for (int r = 0; r < 8; ++r) {
            unsigned char mk = mbase[(size_t)(r + 8 * hi) * SEQ + m0 + lr];
            float p = mk ? 0.f : __expf(c[r] - rowmax[r]);
            st[(r + 8 * hi) * 17 + lr] = p;
        }
        v2f pf[4];
#pragma unroll
        for (int kk2 = 0; kk2 < 4; ++kk2) {
            int k0 = 4 * kk2 + 2 * hi;
            v2f pa;
            pa.x = st[lr * 17 + k0];
            pa.y = st[lr * 17 + k0 + 1];
            pf[kk2] = pa;
        }
        const float* vrow = vbat + (size_t)m0 * CH;
#pragma unroll
        for (int nt = 0; nt < 8; ++nt) {
            int n0 = nt * 16;
#pragma unroll
            for (int kk2 = 0; kk2 < 4; ++kk2) {
                int k0 = 4 * kk2 + 2 * hi;
                v2f vb;
                vb.x = vrow[(size_t)(k0 + 0) * CH + n0 + lr];
                vb.y = vrow[(size_t)(k0 + 1) * CH + n0 + lr];
                acc[nt] = wmma4(pf[kk2], vb, acc[nt]);
            }
        }
    }

    // ---- epilogue: normalize by rowsum, LayerNorm over W=128, + residual ----
#pragma unroll
    for (int r = 0; r < 8; ++r) {
        float inv = rowinv[r];
        float vals[8];
        float s1 = 0.f, s2 = 0.f;
#pragma unroll
        for (int nt = 0; nt < 8; ++nt) {
            float x = acc[nt][r] * inv;
            vals[nt] = x;
            s1 += x;
            s2 += x * x;
        }
#pragma unroll
        for (int d = 1; d < 16; d <<= 1) {
            s1 += __shfl_xor(s1, d);
            s2 += __shfl_xor(s2, d);
        }
        float mu   = s1 * (1.f / 128.f);
        float var  = s2 * (1.f / 128.f) - mu * mu;
        float rstd = rsqrtf(var + 1e-5f);
        int R = r0 + r + 8 * hi;
        const float* x1row = x1 + ((size_t)b * SEQ + R) * CH;
        float* orow = Out + ((size_t)b * SEQ + R) * CH;
#pragma unroll
        for (int nt = 0; nt < 8; ++nt) {
            int col = nt * 16 + lr;
            orow[col] = (vals[nt] - mu) * rstd * ln_g[col] + ln_b[col] + x1row[col];
        }
    }
}

// ---------------------------------------------------------------------------
extern "C" void kernel_launch(void* const* d_in, const int* in_sizes, int n_in,
                              void* d_out, int out_size, void* d_ws, size_t ws_size,
                              hipStream_t stream) {
    (void)in_sizes; (void)n_in; (void)out_size; (void)ws_size;
    const float* x1 = (const float*)d_in[0];
    const float* x2 = (const float*)d_in[1];
    const unsigned char* mask = (const unsigned char*)d_in[2];   // jax bool = 1 byte
    const float* q_w = (const float*)d_in[3];
    const float* q_b = (const float*)d_in[4];
    const float* k_w = (const float*)d_in[5];
    const float* k_b = (const float*)d_in[6];
    const float* v_w = (const float*)d_in[7];
    const float* v_b = (const float*)d_in[8];
    const float* ln_g = (const float*)d_in[9];
    const float* ln_b = (const float*)d_in[10];
    float* out = (float*)d_out;

    float* ws   = (float*)d_ws;
    float* T    = ws;               // 3*16384
    float* tb   = ws + 49152;       // 3*128
    float* beff = ws + 49536;       // 3*128
    float* Wt   = ws + 49920;       // 3*16384
    float* Qb   = ws + 99072;       // 16*2048*128
    float* Kb   = Qb + 4194304;
    float* Vb   = Kb + 4194304;     // total ~50.7 MB

    collapse1_kernel<<<dim3(128, 3, 1), 128, 0, stream>>>(q_w, k_w, v_w, q_b, k_b, v_b, T, tb);
    collapse2_kernel<<<dim3(128, 3, 1), 128, 0, stream>>>(q_w, k_w, v_w, q_b, k_b, v_b, T, tb, Wt, beff);

    // 32768 rows / (16 rows/wave * 4 waves/block) = 512 blocks
    proj_kernel<<<512, 128, 0, stream>>>(x1, Wt,         beff,       Qb);
    proj_kernel<<<512, 128, 0, stream>>>(x2, Wt + 16384, beff + 128, Kb);
    proj_kernel<<<512, 128, 0, stream>>>(x2, Wt + 32768, beff + 256, Vb);

    // 2048 query-tiles/batch / 16 rows = 128 waves / 4 per block = 32 blocks x 16 batches
    attn_kernel<<<dim3(32, 16, 1), 128, 0, stream>>>(Qb, Kb, Vb, mask, x1, ln_g, ln_b, out);
}